// GatedAttention_55233279426706
// MI455X (gfx1250) — compile-verified
//
#include <hip/hip_runtime.h>
#include <cstddef>

#define DIM 256
#define LOG2E 1.4426950408889634f

typedef float v2f __attribute__((ext_vector_type(2)));
typedef float v8f __attribute__((ext_vector_type(8)));

// ---------------------------------------------------------------------------
// Kernel 0: qb = dot(query, wq) + gate_b   (single wave)
// ---------------------------------------------------------------------------
__global__ __launch_bounds__(32) void ga_prep(const float* __restrict__ query,
                                              const float* __restrict__ gate_w,
                                              const float* __restrict__ gate_b,
                                              float* __restrict__ qb) {
  int lane = threadIdx.x;  // 0..31
  const float4* q4 = reinterpret_cast<const float4*>(query);
  const float4* w4 = reinterpret_cast<const float4*>(gate_w);  // wq = gate_w[0:256]
  float d = 0.0f;
#pragma unroll
  for (int j = 0; j < 2; ++j) {
    float4 a = q4[lane * 2 + j];
    float4 b = w4[lane * 2 + j];
    d = fmaf(a.x, b.x, d); d = fmaf(a.y, b.y, d);
    d = fmaf(a.z, b.z, d); d = fmaf(a.w, b.w, d);
  }
#pragma unroll
  for (int m = 16; m >= 1; m >>= 1) d += __shfl_xor(d, m, 32);
  if (lane == 0) qb[0] = d + gate_b[0];
}

// ---------------------------------------------------------------------------
// Kernel 1: fused streaming pass (one read of M):
//   z_i = (M_i . wm + qb) * weights_i
//   partial s_b   = sum exp(z_i)
//   partial acc_b = sum exp(z_i) * M_i
// Wave layout: 4 rows per iteration, 8 lanes per row, 32 columns per lane.
// GUARD=false specialization (exact tiling) has no bounds checks in the
// bandwidth-critical loop.
// ---------------------------------------------------------------------------
template <bool GUARD>
__global__ __launch_bounds__(256) void ga_pass1(const float* __restrict__ M,
                                                const float* __restrict__ weights,
                                                const float* __restrict__ wm,
                                                const float* __restrict__ qb,
                                                float* __restrict__ accOut,
                                                float* __restrict__ sOut,
                                                int rowsPerBlock, int N) {
  __shared__ float s_wm[DIM];
  __shared__ float s_acc[8][DIM];
  __shared__ float s_sums[8];

  const int tid = threadIdx.x;
  s_wm[tid] = wm[tid];
  __syncthreads();

  const int wave = tid >> 5;
  const int lane = tid & 31;
  const int rg   = lane >> 3;   // row group within wave (0..3)
  const int lpos = lane & 7;    // lane position within row (0..7)

  const int rowsPerWave = rowsPerBlock >> 3;
  const int iters       = rowsPerWave >> 2;
  const int rowBase     = blockIdx.x * rowsPerBlock + wave * rowsPerWave + rg;

  float4 wmf[8];
#pragma unroll
  for (int j = 0; j < 8; ++j)
    wmf[j] = *reinterpret_cast<const float4*>(&s_wm[lpos * 4 + 32 * j]);

  const float qbv = qb[0];

  float4 acc[8];
#pragma unroll
  for (int j = 0; j < 8; ++j) acc[j] = make_float4(0.f, 0.f, 0.f, 0.f);
  float sl = 0.0f;

  for (int it = 0; it < iters; ++it) {
    const int row = rowBase + it * 4;
    const bool valid = GUARD ? (row < N) : true;
    const float* rp = M + (size_t)row * DIM + lpos * 4;

    // Unconditional prefetch of the next iteration's line, address clamped
    // in-bounds (WGP-scope prefetch is non-speculative).
    {
      int pr = row + 4;
      if (pr > N - 1) pr = N - 1;
      __builtin_prefetch(M + (size_t)pr * DIM + lpos * 4, 0, 1);
    }

    float4 v[8];
#pragma unroll
    for (int j = 0; j < 8; ++j) {
      if (valid) v[j] = *reinterpret_cast<const float4*>(rp + 32 * j);
      else       v[j] = make_float4(0.f, 0.f, 0.f, 0.f);
    }

    float d = 0.0f;
#pragma unroll
    for (int j = 0; j < 8; ++j) {
      d = fmaf(v[j].x, wmf[j].x, d);
      d = fmaf(v[j].y, wmf[j].y, d);
      d = fmaf(v[j].z, wmf[j].z, d);
      d = fmaf(v[j].w, wmf[j].w, d);
    }
    // reduce partial dot across the 8 lanes of this row
    d += __shfl_xor(d, 1, 32);
    d += __shfl_xor(d, 2, 32);
    d += __shfl_xor(d, 4, 32);

    float w = valid ? weights[row] : 0.0f;
    float e = valid ? exp2f((d + qbv) * w * LOG2E) : 0.0f;

    sl += e;
#pragma unroll
    for (int j = 0; j < 8; ++j) {
      acc[j].x = fmaf(e, v[j].x, acc[j].x);
      acc[j].y = fmaf(e, v[j].y, acc[j].y);
      acc[j].z = fmaf(e, v[j].z, acc[j].z);
      acc[j].w = fmaf(e, v[j].w, acc[j].w);
    }
  }

  // Combine the 4 row groups of this wave (lanes differing in bits 3,4).
#pragma unroll
  for (int j = 0; j < 8; ++j) {
    acc[j].x += __shfl_xor(acc[j].x, 8, 32);  acc[j].x += __shfl_xor(acc[j].x, 16, 32);
    acc[j].y += __shfl_xor(acc[j].y, 8, 32);  acc[j].y += __shfl_xor(acc[j].y, 16, 32);
    acc[j].z += __shfl_xor(acc[j].z, 8, 32);  acc[j].z += __shfl_xor(acc[j].z, 16, 32);
    acc[j].w += __shfl_xor(acc[j].w, 8, 32);  acc[j].w += __shfl_xor(acc[j].w, 16, 32);
  }
#pragma unroll
  for (int m = 16; m >= 1; m >>= 1) sl += __shfl_xor(sl, m, 32);

  if (lane < 8) {
#pragma unroll
    for (int j = 0; j < 8; ++j)
      *reinterpret_cast<float4*>(&s_acc[wave][lpos * 4 + 32 * j]) = acc[j];
  }
  if (lane == 0) s_sums[wave] = sl;
  __syncthreads();

  // Deterministic block combine: each thread owns one column.
  float a = 0.0f;
#pragma unroll
  for (int w = 0; w < 8; ++w) a += s_acc[w][tid];
  accOut[(size_t)blockIdx.x * DIM + tid] = a;

  if (tid == 0) {
    float st = 0.0f;
#pragma unroll
    for (int w = 0; w < 8; ++w) st += s_sums[w];
    sOut[blockIdx.x] = st * 0.125f;  // each row's e was counted by 8 lanes
  }
}

// ---------------------------------------------------------------------------
// Kernel 2: finalize.  out[c] = (sum_b acc_b[c]) / (sum_b s_b)
// The column sum is a ones-vector GEMV done with V_WMMA_F32_16X16X4_F32:
//   D = A x B + C  with A = all-ones 16x4  ->  every row of D is the K-sum of B.
// All loads in the WMMA loop are unconditional; the <=3-row remainder is
// added with scalar VALU afterwards.
// ---------------------------------------------------------------------------
__global__ __launch_bounds__(256) void ga_finalize(const float* __restrict__ acc,
                                                   const float* __restrict__ s,
                                                   int blocks,
                                                   float* __restrict__ out) {
  __shared__ float swave[8];
  const int tid  = threadIdx.x;
  const int wave = tid >> 5;
  const int lane = tid & 31;

  // ---- total softmax denominator (deterministic tree) ----
  float sl = 0.0f;
  for (int b = tid; b < blocks; b += 256) sl += s[b];
#pragma unroll
  for (int m = 16; m >= 1; m >>= 1) sl += __shfl_xor(sl, m, 32);
  if (lane == 0) swave[wave] = sl;
  __syncthreads();
  float Stot = 0.0f;
#pragma unroll
  for (int w = 0; w < 8; ++w) Stot += swave[w];
  const float invS = 1.0f / Stot;

  // ---- column sums via f32 WMMA ----
  const int colHalf = lane & 15;
  const int koff    = (lane >> 4) ? 2 : 0;
  const v2f ones    = {1.0f, 1.0f};
  const int groups  = blocks >> 2;      // full groups of 4 partial rows
  const int rem0    = groups << 2;      // first remainder row

#pragma unroll
  for (int half = 0; half < 2; ++half) {
    const int n0  = wave * 32 + half * 16;   // 16-column chunk
    const int col = n0 + colHalf;
    v8f c = {};
    const float* p = acc + (size_t)koff * DIM + col;
    for (int g = 0; g < groups; ++g) {
      v2f bf;
      bf.x = p[0];
      bf.y = p[DIM];
      p += 4 * DIM;
      c = __builtin_amdgcn_wmma_f32_16x16x4_f32(false, ones, false, bf,
                                                (short)0, c, false, false);
    }
    // Remainder rows (< 4): plain scalar adds into the M=0 row result.
    float extra = 0.0f;
    for (int b = rem0; b < blocks; ++b) extra += acc[(size_t)b * DIM + col];
    if (lane < 16) out[col] = (c[0] + extra) * invS;
  }
}

// ---------------------------------------------------------------------------
extern "C" void kernel_launch(void* const* d_in, const int* in_sizes, int n_in,
                              void* d_out, int out_size, void* d_ws, size_t ws_size,
                              hipStream_t stream) {
  const float* query   = (const float*)d_in[0];
  const float* M       = (const float*)d_in[1];
  const float* weights = (const float*)d_in[2];
  const float* gate_w  = (const float*)d_in[3];
  const float* gate_b  = (const float*)d_in[4];
  float* out = (float*)d_out;
  float* ws  = (float*)d_ws;

  const int N = in_sizes[2];  // number of memory rows

  // Workspace layout: [0]=qb, [16 .. 16+blocks*256) = acc partials, then s partials.
  long wsFloats = (long)(ws_size / sizeof(float));
  long maxB = (wsFloats - 16) / (DIM + 1);
  int blocks = 512;
  if (maxB < blocks) blocks = (int)maxB;
  if (blocks < 1) blocks = 1;

  int rowsPerBlock = ((N + blocks - 1) / blocks + 31) & ~31;  // multiple of 32

  float* qb   = ws;
  float* accP = ws + 16;
  float* sP   = accP + (size_t)blocks * DIM;

  ga_prep<<<1, 32, 0, stream>>>(query, gate_w, gate_b, qb);

  const bool exact = ((long)blocks * (long)rowsPerBlock == (long)N);
  if (exact) {
    ga_pass1<false><<<blocks, 256, 0, stream>>>(M, weights, gate_w + DIM, qb,
                                                accP, sP, rowsPerBlock, N);
  } else {
    ga_pass1<true><<<blocks, 256, 0, stream>>>(M, weights, gate_w + DIM, qb,
                                               accP, sP, rowsPerBlock, N);
  }
  ga_finalize<<<1, 256, 0, stream>>>(accP, sP, blocks, out);
}